// SDN_55018531062234
// MI455X (gfx1250) — compile-verified
//
#include <hip/hip_runtime.h>

// Problem constants (match reference)
#define HN  6
#define BN  512
#define DN  2048
#define CN  1000
#define TAU 0.5f

typedef __attribute__((ext_vector_type(2))) float v2f;
typedef __attribute__((ext_vector_type(8))) float v8f;

// ---------------------------------------------------------------------------
// Kernel 1: logits[h,b,c] = feats[h]@W[h] + bias[h]  via V_WMMA_F32_16X16X4_F32
// Wave = 64(M) x 16(N) strip, 4 v8f accumulators, K loop step 4, unroll 4.
// Waves: HN * (512/64=8) * (1008/16=63) = 3024  -> 378 blocks x 8 waves.
// Out-of-range columns (n >= 1000) are CLAMPED at load time (always-legal
// unconditional loads, no exec-mask dance) and masked only at store time:
// column n is owned end-to-end by lane (n&15), so clamped lanes only produce
// garbage D columns that are never written.
// ---------------------------------------------------------------------------
__global__ __launch_bounds__(256) void sdn_gemm_bias(
    const float* __restrict__ feats, const float* __restrict__ Wm,
    const float* __restrict__ bias, float* __restrict__ logits) {
  const int wid  = blockIdx.x * 8 + (threadIdx.x >> 5);
  const int lane = threadIdx.x & 31;

  const int h  = wid / (8 * 63);
  const int r  = wid % (8 * 63);
  const int mt = r / 63;
  const int nt = r % 63;
  const int mbase = mt * 64;
  const int nbase = nt * 16;

  const int l15   = lane & 15;
  const int khalf = lane >> 4;          // 0: K=k,k+1   1: K=k+2,k+3

  const float* __restrict__ Ah = feats + (size_t)h * BN * DN;
  const float* __restrict__ Wh = Wm    + (size_t)h * DN * CN;

  const int  n      = nbase + l15;
  const bool nvalid = (n < CN);
  const int  nc     = nvalid ? n : (CN - 1);   // clamped, always in-bounds

  // 32-bit element offsets from the (wave-uniform) head base pointers.
  const int a0o = (mbase +  0 + l15) * DN + 2 * khalf;
  const int a1o = (mbase + 16 + l15) * DN + 2 * khalf;
  const int a2o = (mbase + 32 + l15) * DN + 2 * khalf;
  const int a3o = (mbase + 48 + l15) * DN + 2 * khalf;
  const int b0o = (2 * khalf) * CN + nc;       // element (2*khalf, nc)

  v8f acc0 = {}, acc1 = {}, acc2 = {}, acc3 = {};

#pragma unroll 4
  for (int k = 0; k < DN; k += 4) {
    v2f av0 = *(const v2f*)(Ah + a0o + k);
    v2f av1 = *(const v2f*)(Ah + a1o + k);
    v2f av2 = *(const v2f*)(Ah + a2o + k);
    v2f av3 = *(const v2f*)(Ah + a3o + k);
    v2f bv;
    bv.x = Wh[b0o + k * CN];
    bv.y = Wh[b0o + (k + 1) * CN];

    acc0 = __builtin_amdgcn_wmma_f32_16x16x4_f32(false, av0, false, bv, (short)0, acc0, false, false);
    acc1 = __builtin_amdgcn_wmma_f32_16x16x4_f32(false, av1, false, bv, (short)0, acc1, false, false);
    acc2 = __builtin_amdgcn_wmma_f32_16x16x4_f32(false, av2, false, bv, (short)0, acc2, false, false);
    acc3 = __builtin_amdgcn_wmma_f32_16x16x4_f32(false, av3, false, bv, (short)0, acc3, false, false);
  }

  const float bv = bias[h * CN + nc];

  // C/D layout: VGPR i -> M = mbase+sub*16 + khalf*8 + i, N = nbase + l15
  if (nvalid) {
#pragma unroll
    for (int i = 0; i < 8; ++i) {
      int row0 = mbase +  0 + khalf * 8 + i;
      int row1 = mbase + 16 + khalf * 8 + i;
      int row2 = mbase + 32 + khalf * 8 + i;
      int row3 = mbase + 48 + khalf * 8 + i;
      logits[(size_t)(h * BN + row0) * CN + n] = acc0[i] + bv;
      logits[(size_t)(h * BN + row1) * CN + n] = acc1[i] + bv;
      logits[(size_t)(h * BN + row2) * CN + n] = acc2[i] + bv;
      logits[(size_t)(h * BN + row3) * CN + n] = acc3[i] + bv;
    }
  }
}

// ---------------------------------------------------------------------------
// Kernel 2: per-row (h*BN+b) softmax stats: lse, conf = 1/sum(exp(x-max)), argmax
// One wave per row; 4 waves / 128-thread block; 3072 rows -> 768 blocks.
// ---------------------------------------------------------------------------
__global__ __launch_bounds__(128) void sdn_softmax_stats(
    const float* __restrict__ logits, float* __restrict__ lse,
    float* __restrict__ conf, int* __restrict__ amax) {
  const int r    = blockIdx.x * 4 + (threadIdx.x >> 5);
  const int lane = threadIdx.x & 31;
  const float* base = logits + (size_t)r * CN;

  float m = -3.402823466e38f;
  int   idx = 0;
  for (int j = lane; j < CN; j += 32) {
    float v = base[j];
    if (v > m) { m = v; idx = j; }
  }
#pragma unroll
  for (int off = 16; off > 0; off >>= 1) {
    float om = __shfl_xor(m, off, 32);
    int   oi = __shfl_xor(idx, off, 32);
    if (om > m || (om == m && oi < idx)) { m = om; idx = oi; }
  }

  float s = 0.0f;
  for (int j = lane; j < CN; j += 32) s += expf(base[j] - m);
#pragma unroll
  for (int off = 16; off > 0; off >>= 1) s += __shfl_xor(s, off, 32);

  if (lane == 0) {
    lse[r]  = m + logf(s);
    conf[r] = 1.0f / s;          // exp(max - lse)
    amax[r] = idx;
  }
}

// ---------------------------------------------------------------------------
// Kernel 3: routing + gather + per-sample loss/acc. One wave per sample.
// 512 samples -> 32 blocks x 16 waves (512 threads).
// ---------------------------------------------------------------------------
__global__ __launch_bounds__(512) void sdn_route(
    const float* __restrict__ logits, const float* __restrict__ lse,
    const float* __restrict__ conf, const int* __restrict__ amax,
    const int* __restrict__ y_true, float* __restrict__ outp,
    int* __restrict__ eidx, float* __restrict__ lossA, float* __restrict__ corrA) {
  const int s    = blockIdx.x * 16 + (threadIdx.x >> 5);
  const int lane = threadIdx.x & 31;

  int e = 0;
  if (lane == 0) {
    int first = -1, best = 0;
    float bc = -1.0f;
    for (int h = 0; h < HN; ++h) {
      float c = conf[h * BN + s];
      if (first < 0 && c >= TAU) first = h;
      if (c > bc) { bc = c; best = h; }   // strict '>' keeps first occurrence
    }
    e = (first >= 0) ? first : best;
  }
  e = __shfl(e, 0, 32);

  const int row = e * BN + s;
  const float4* src = (const float4*)(logits + (size_t)row * CN);
  float4*       dst = (float4*)(outp + (size_t)s * CN);
  for (int i = lane; i < CN / 4; i += 32) dst[i] = src[i];

  if (lane == 0) {
    int y = y_true[s];
    eidx[s]  = e;
    lossA[s] = -(logits[(size_t)row * CN + y] - lse[row]);
    corrA[s] = (amax[row] == y) ? 1.0f : 0.0f;
  }
}

// ---------------------------------------------------------------------------
// Kernel 4: reduce per-sample loss/correct -> mean loss / accuracy.
// ---------------------------------------------------------------------------
__global__ __launch_bounds__(512) void sdn_finalize(
    const float* __restrict__ lossA, const float* __restrict__ corrA,
    float* __restrict__ outTail) {
  __shared__ float sl[512];
  __shared__ float sc[512];
  const int t = threadIdx.x;
  sl[t] = lossA[t];
  sc[t] = corrA[t];
  __syncthreads();
#pragma unroll
  for (int off = 256; off > 0; off >>= 1) {
    if (t < off) { sl[t] += sl[t + off]; sc[t] += sc[t + off]; }
    __syncthreads();
  }
  if (t == 0) {
    outTail[0] = sl[0] / (float)BN;   // loss
    outTail[1] = sc[0] / (float)BN;   // acc
  }
}

// ---------------------------------------------------------------------------
extern "C" void kernel_launch(void* const* d_in, const int* in_sizes, int n_in,
                              void* d_out, int out_size, void* d_ws, size_t ws_size,
                              hipStream_t stream) {
  const float* feats = (const float*)d_in[0];   // [H,B,D]
  const float* Wm    = (const float*)d_in[1];   // [H,D,C]
  const float* bias  = (const float*)d_in[2];   // [H,C]
  const int*   ytrue = (const int*)d_in[3];     // [B]

  // Output layout: [B*C outputs | B exit_idx (i32) | loss | acc]
  float* outp    = (float*)d_out;
  int*   eidx    = (int*)d_out + (size_t)BN * CN;
  float* outTail = (float*)d_out + (size_t)BN * CN + BN;

  // Workspace layout (floats): logits[H*B*C], lse[H*B], conf[H*B], amax[H*B](int),
  //                            lossA[B], corrA[B]
  float* logits = (float*)d_ws;
  float* lse    = logits + (size_t)HN * BN * CN;
  float* conf   = lse + HN * BN;
  int*   amax   = (int*)(conf + HN * BN);
  float* lossA  = (float*)(amax + HN * BN);
  float* corrA  = lossA + BN;

  // 1) GEMM + bias: 3024 waves = 378 blocks * 8 waves
  sdn_gemm_bias<<<378, 256, 0, stream>>>(feats, Wm, bias, logits);
  // 2) softmax stats: 3072 rows, 4 waves/block
  sdn_softmax_stats<<<768, 128, 0, stream>>>(logits, lse, conf, amax);
  // 3) routing + gather: 512 samples, 16 waves/block
  sdn_route<<<32, 512, 0, stream>>>(logits, lse, conf, amax, ytrue,
                                    outp, eidx, lossA, corrA);
  // 4) final mean loss / accuracy
  sdn_finalize<<<1, 512, 0, stream>>>(lossA, corrA, outTail);
}